// HGNN_conv_41644002902021
// MI455X (gfx1250) — compile-verified
//
#include <hip/hip_runtime.h>

// HGNN conv: out = 0.5 * D_e ∘ (MT^T @ (D_v ∘ (MT @ (x @ W))))
// N=16384 nodes, E=8192 edges, IN_FT=256, OUT_FT=128, all fp32.
// Three V_WMMA_F32_16X16X4_F32 GEMM kernels:
//   K1: yT[f][n]  = (x @ W)^T           (workspace, 8 MB, L2-resident)
//   K2: eyT[f][e] = (D_v ∘ (MT @ y))^T  (workspace, 4 MB, L2-resident)
//   K3: out[n][f] = 0.5 * D_e ∘ (MT^T @ ey)
// MT (512 MB) is streamed once per GEMM with non-temporal loads.
// K2/K3 pair two waves per MT tile (even/odd wave in the same workgroup,
// f-halves) so the duplicated MT frags hit the WGP$ while wave parallelism
// doubles (K2: 1024 waves, K3: 2048 waves) for latency hiding.

#define NN 16384   // nodes
#define NE 8192    // edges
#define CIN 256    // IN_FT
#define FOUT 128   // OUT_FT

typedef float v2f __attribute__((ext_vector_type(2)));
typedef float v8f __attribute__((ext_vector_type(8)));

__device__ __forceinline__ v8f wmma_f32(v2f a, v2f b, v8f c) {
  // D = A(16x4 f32) x B(4x16 f32) + C(16x16 f32)
  return __builtin_amdgcn_wmma_f32_16x16x4_f32(
      false, a, false, b, (short)0, c, false, false);
}

__device__ __forceinline__ v2f ld2(const float* p) {
  return *(const v2f*)p;
}
__device__ __forceinline__ v2f ld2_nt(const float* p) {
  return __builtin_nontemporal_load((const v2f*)p);
}
__device__ __forceinline__ float ld_nt(const float* p) {
  return __builtin_nontemporal_load(p);
}

// ---------------------------------------------------------------------------
// K1: yT[f][n] = sum_c x[n][c] * W[c][f]      (A = W^T frags, B = x^T frags)
// Wave tile: 128 f (8 M-tiles) x 16 n.  Grid: 1024 waves.
// ---------------------------------------------------------------------------
__global__ void __launch_bounds__(256) hgnn_k1_xw(
    const float* __restrict__ x, const float* __restrict__ w,
    float* __restrict__ yT) {
  const int lane = threadIdx.x & 31;
  const int wave = threadIdx.x >> 5;
  const int lm = lane & 15;   // row/col within tile
  const int kh = lane >> 4;   // K-half: 0 -> K=k,k+1   1 -> K=k+2,k+3
  const int n0 = (blockIdx.x * 8 + wave) * 16;

  v8f acc[8];
  const v8f vzero = {0.f, 0.f, 0.f, 0.f, 0.f, 0.f, 0.f, 0.f};
#pragma unroll
  for (int t = 0; t < 8; ++t) acc[t] = vzero;

  const float* xrow = x + (size_t)(n0 + lm) * CIN + 2 * kh;   // B frag base
  const float* wbase = w + (size_t)(2 * kh) * FOUT + lm;      // A frag base

  for (int k = 0; k < CIN; k += 4) {
    v2f b = ld2(xrow + k);
#pragma unroll
    for (int t = 0; t < 8; ++t) {
      v2f a;
      a.x = wbase[k * FOUT + t * 16];          // W[k+2kh][t*16+lm]
      a.y = wbase[k * FOUT + FOUT + t * 16];   // W[k+2kh+1][t*16+lm]
      acc[t] = wmma_f32(a, b, acc[t]);
    }
  }
#pragma unroll
  for (int t = 0; t < 8; ++t)
#pragma unroll
    for (int r = 0; r < 8; ++r)
      yT[(size_t)(t * 16 + r + 8 * kh) * NN + n0 + lm] = acc[t][r];
}

// ---------------------------------------------------------------------------
// K2: eyT[f][e] = D_v[e] * sum_n MT[e][n] * yT[f][n]
// A = yT (b64 frags), B = MT (b64 NT frags; lane lm streams MT row e0+lm).
// Wave tile: 64 f (4 tiles, f-half = wave&1) x 16 e.
// Wave pair (2j, 2j+1) shares one e-tile -> MT frags shared via WGP$.
// Grid: 1024 waves (128 blocks x 8 waves), k-loop over 16384.
// ---------------------------------------------------------------------------
__global__ void __launch_bounds__(256) hgnn_k2_mty(
    const float* __restrict__ MT, const float* __restrict__ yT,
    const float* __restrict__ Dv, float* __restrict__ eyT) {
  const int lane = threadIdx.x & 31;
  const int wave = threadIdx.x >> 5;
  const int lm = lane & 15;
  const int kh = lane >> 4;
  const int fh = (wave & 1) * 64;                     // f-half: 0 or 64
  const int e0 = (blockIdx.x * 4 + (wave >> 1)) * 16; // e-tile

  v8f acc[4];
  const v8f vzero = {0.f, 0.f, 0.f, 0.f, 0.f, 0.f, 0.f, 0.f};
#pragma unroll
  for (int t = 0; t < 4; ++t) acc[t] = vzero;

  // B[K=n][N=e] = MT[e][n]: lane lm owns MT row e0+lm, contiguous K pair
  const float* mrow = MT + (size_t)(e0 + lm) * NN + 2 * kh;
  // A[M=f][K=n] = yT[f][n]: contiguous K pair
  const float* ybase = yT + (size_t)(fh + lm) * NN + 2 * kh;

#pragma unroll 2
  for (int k = 0; k < NN; k += 4) {
    v2f b = ld2_nt(mrow + k);
#pragma unroll
    for (int t = 0; t < 4; ++t) {
      v2f a = ld2(ybase + (size_t)t * 16 * NN + k);
      acc[t] = wmma_f32(a, b, acc[t]);
    }
  }
  const float s = Dv[e0 + lm];  // column scaling: same for all rows in lane
#pragma unroll
  for (int t = 0; t < 4; ++t)
#pragma unroll
    for (int r = 0; r < 8; ++r)
      eyT[(size_t)(fh + t * 16 + r + 8 * kh) * NE + e0 + lm] = acc[t][r] * s;
}

// ---------------------------------------------------------------------------
// K3: out[n][f] = 0.5 * D_e[n] * sum_e MT[e][n] * eyT[f][e]
// A = MT^T frags (2 x b32 NT loads, 2x64B segments of MT rows),
// B = eyT (b64 frags, L2/WGP$-resident).
// Wave tile: 16 n x 64 f (4 tiles, f-half = wave&1).
// Wave pair (2j, 2j+1) shares one n-tile -> MT frags shared via WGP$.
// Grid: 2048 waves (256 blocks x 8 waves), k-loop over 8192.
// ---------------------------------------------------------------------------
__global__ void __launch_bounds__(256) hgnn_k3_mttey(
    const float* __restrict__ MT, const float* __restrict__ eyT,
    const float* __restrict__ De, float* __restrict__ out) {
  const int lane = threadIdx.x & 31;
  const int wave = threadIdx.x >> 5;
  const int lm = lane & 15;
  const int kh = lane >> 4;
  const int fh = (wave & 1) * 64;                     // f-half: 0 or 64
  const int n0 = (blockIdx.x * 4 + (wave >> 1)) * 16; // n-tile

  v8f acc[4];
  const v8f vzero = {0.f, 0.f, 0.f, 0.f, 0.f, 0.f, 0.f, 0.f};
#pragma unroll
  for (int t = 0; t < 4; ++t) acc[t] = vzero;

  // A[M=n][K=e] = MT[e][n]: v0 from row k+2kh, v1 from row k+2kh+1
  const float* mcol = MT + (size_t)(2 * kh) * NN + n0 + lm;
  // B[K=e][N=f] = eyT[f][e]: contiguous K pair per lane
  const float* ebase = eyT + (size_t)(fh + lm) * NE + 2 * kh;

#pragma unroll 2
  for (int k = 0; k < NE; k += 4) {
    v2f a;
    a.x = ld_nt(mcol + (size_t)k * NN);         // MT[k+2kh][n0+lm]
    a.y = ld_nt(mcol + (size_t)k * NN + NN);    // MT[k+2kh+1][n0+lm]
#pragma unroll
    for (int t = 0; t < 4; ++t) {
      v2f b = ld2(ebase + (size_t)t * 16 * NE + k);
      acc[t] = wmma_f32(a, b, acc[t]);
    }
  }
#pragma unroll
  for (int r = 0; r < 8; ++r) {
    const int row = n0 + r + 8 * kh;
    const float s = 0.5f * De[row];  // row scaling
#pragma unroll
    for (int t = 0; t < 4; ++t)
      out[(size_t)row * FOUT + fh + t * 16 + lm] = acc[t][r] * s;
  }
}

// ---------------------------------------------------------------------------
extern "C" void kernel_launch(void* const* d_in, const int* in_sizes, int n_in,
                              void* d_out, int out_size, void* d_ws,
                              size_t ws_size, hipStream_t stream) {
  (void)in_sizes; (void)n_in; (void)out_size; (void)ws_size;
  const float* x  = (const float*)d_in[0];  // [N, 256]
  const float* w  = (const float*)d_in[1];  // [256, 128]
  const float* MT = (const float*)d_in[2];  // [E, N]
  const float* Dv = (const float*)d_in[3];  // [E]
  const float* De = (const float*)d_in[4];  // [N]
  float* out = (float*)d_out;               // [N, 128]

  float* yT  = (float*)d_ws;                 // [128][16384]  8 MB
  float* eyT = yT + (size_t)FOUT * NN;       // [128][8192]   4 MB

  // K1: 1024 n-tiles, 8 waves/block
  hgnn_k1_xw<<<NN / 16 / 8, 256, 0, stream>>>(x, w, yT);
  // K2: 512 e-tiles x 2 f-halves = 1024 waves, 4 e-tiles/block
  hgnn_k2_mty<<<NE / 16 / 4, 256, 0, stream>>>(MT, yT, Dv, eyT);
  // K3: 1024 n-tiles x 2 f-halves = 2048 waves, 4 n-tiles/block
  hgnn_k3_mttey<<<NN / 16 / 4, 256, 0, stream>>>(MT, eyT, De, out);
}